// GatedHebbianUnit_19172734009491
// MI455X (gfx1250) — compile-verified
//
#include <hip/hip_runtime.h>
#include <cmath>

typedef __attribute__((ext_vector_type(2))) float v2f;
typedef __attribute__((ext_vector_type(4))) float v4f;
typedef __attribute__((ext_vector_type(8))) float v8f;

#define P_ 8
#define B_ 256
#define N_ 512
#define K_ 64

__device__ __forceinline__ v8f wmma4(v2f a, v2f b, v8f c) {
  return __builtin_amdgcn_wmma_f32_16x16x4_f32(false, a, false, b, (short)0, c,
                                               false, false);
}

// Phase 1 inner loop: A rows stream from global (per-lane b64, element offset
// pre-folded), B chunks broadcast from LDS (per-lane b64). Branch-free body.
template <bool BWD, bool FWD>
__device__ __forceinline__ void phase1_loop(const float* __restrict__ Arow,
                                            const float* sb, const float* sf,
                                            int n_begin, v8f& accb, v8f& accf) {
  for (int n0 = n_begin; n0 < n_begin + N_ / 2; n0 += 16) {
    v2f a[4];
#pragma unroll
    for (int j = 0; j < 4; ++j)
      a[j] = __builtin_nontemporal_load((const v2f*)(Arow + n0 + 4 * j));
    v2f bb[4], bf[4];
#pragma unroll
    for (int j = 0; j < 4; ++j) {
      if (BWD) bb[j] = *(const v2f*)(sb + n0 + 4 * j);
      if (FWD) bf[j] = *(const v2f*)(sf + n0 + 4 * j);
    }
#pragma unroll
    for (int j = 0; j < 4; ++j) {
      if (BWD) accb = wmma4(a[j], bb[j], accb);
      if (FWD) accf = wmma4(a[j], bf[j], accf);
    }
  }
}

// Phase 2 tile: K=64 reduction, 16 chunks; all 16 A loads issued up front.
template <bool BWD, bool FWD>
__device__ __forceinline__ void phase2_tile(const float* __restrict__ Arow,
                                            const float* sb, const float* sf,
                                            v8f& accb, v8f& accf) {
  v2f a[16];
#pragma unroll
  for (int j = 0; j < 16; ++j)
    a[j] = __builtin_nontemporal_load((const v2f*)(Arow + 4 * j));
#pragma unroll
  for (int j = 0; j < 16; ++j) {
    if (BWD) accb = wmma4(a[j], *(const v2f*)(sb + 4 * j), accb);
    if (FWD) accf = wmma4(a[j], *(const v2f*)(sf + 4 * j), accf);
  }
}

// Heavy path (needs matrix streaming), specialized on block-uniform flags.
template <bool BWD, bool FWD>
__device__ __forceinline__ void heavy_path(
    const float* __restrict__ WL, const float* __restrict__ WR, int pb, int b,
    int tid, float* __restrict__ v_next, float* __restrict__ dWLpb, float g,
    float* s_vtm1, float* s_vt, float (*s_wrx_part)[K_], float (*s_wx_part)[K_],
    float* s_wrx, float* s_wx) {
  const int wave = tid >> 5;
  const int lane = tid & 31;
  const bool hi = lane >= 16;
  const int eoff = hi ? 2 : 0;  // ISA 16x4 f32 A/B layout: hi half-wave = K 2,3
  const size_t mat_off = (size_t)pb * (N_ * K_);

  // ---- Phase 1: WRx = WR(KxN) * v ----
  {
    const int ktile = wave & 3;
    const int half = wave >> 2;
    const float* Arow =
        WR + mat_off + (size_t)(ktile * 16 + (lane & 15)) * N_ + eoff;
    v8f accb = {};
    v8f accf = {};
    phase1_loop<BWD, FWD>(Arow, s_vtm1 + eoff, s_vt + eoff, half * (N_ / 2),
                          accb, accf);
    if (lane == 0 || lane == 16) {  // D: VGPR r = M=r (lane0) / M=r+8 (lane16)
      const int kb = ktile * 16 + (hi ? 8 : 0);
      if (BWD) {
#pragma unroll
        for (int r = 0; r < 8; ++r) s_wrx_part[half][kb + r] = accb[r];
      }
      if (FWD) {
#pragma unroll
        for (int r = 0; r < 8; ++r) s_wx_part[half][kb + r] = accf[r];
      }
    }
  }
  __syncthreads();
  if (tid < K_) {
    s_wrx[tid] = s_wrx_part[0][tid] + s_wrx_part[1][tid];
    s_wx[tid] = s_wx_part[0][tid] + s_wx_part[1][tid];
  }
  __syncthreads();

  // ---- Phase 2: WLWRx / act = WL(NxK) * {WRx, Wx} ----
  for (int t = wave; t < N_ / 16; t += 8) {
    const int n16 = t * 16;
    const float* Arow =
        WL + mat_off + (size_t)(n16 + (lane & 15)) * K_ + eoff;
    v8f accb = {};
    v8f accf = {};
    phase2_tile<BWD, FWD>(Arow, s_wrx + eoff, s_wx + eoff, accb, accf);
    if (lane == 0 || lane == 16) {
      const int mb = n16 + (hi ? 8 : 0);
      if (BWD) {
        v4f o0, o1;
#pragma unroll
        for (int r = 0; r < 4; ++r) o0[r] = g * s_vt[mb + r] - accb[r];
#pragma unroll
        for (int r = 0; r < 4; ++r) o1[r] = g * s_vt[mb + 4 + r] - accb[4 + r];
        __builtin_nontemporal_store(o0, (v4f*)(dWLpb + mb));
        __builtin_nontemporal_store(o1, (v4f*)(dWLpb + mb + 4));
      }
      if (FWD) {
        v4f o0, o1;
#pragma unroll
        for (int r = 0; r < 4; ++r) o0[r] = tanhf(accf[r]);
#pragma unroll
        for (int r = 0; r < 4; ++r) o1[r] = tanhf(accf[4 + r]);
        __builtin_nontemporal_store(o0, (v4f*)(v_next + b * N_ + mb));
        __builtin_nontemporal_store(o1, (v4f*)(v_next + b * N_ + mb + 4));
      }
    }
  }
}

__global__ __launch_bounds__(256, 2) void ghu_fused(
    const float* __restrict__ WL, const float* __restrict__ WR,
    const float* __restrict__ v_tm1, const float* __restrict__ v_t,
    const int* __restrict__ ac_idx, const int* __restrict__ pc_mask,
    float* __restrict__ v_next, float* __restrict__ dWL,
    float* __restrict__ dWR, float g, float inv_n) {
  __shared__ __align__(16) float s_vtm1[N_];
  __shared__ __align__(16) float s_vt[N_];
  __shared__ __align__(16) float s_wrx_part[2][K_];
  __shared__ __align__(16) float s_wx_part[2][K_];
  __shared__ __align__(16) float s_wrx[K_];
  __shared__ __align__(16) float s_wx[K_];

  const int pb = blockIdx.x;  // pb = p*B + b
  const int p = pb >> 8;
  const int b = pb & 255;
  const int tid = threadIdx.x;

  const bool need_fwd = (ac_idx[b] == p);
  const bool need_bwd = (pc_mask[b * P_ + p] != 0);

  const float* vtm1b = v_tm1 + b * N_;
  const float* vtb = v_t + b * N_;
  float* dWLpb = dWL + (size_t)pb * N_;
  float* dWRpb = dWR + (size_t)pb * N_;

  // Phase 0: vectors -> LDS, dWR elementwise, zero-fill unmasked dWL.
  for (int i = tid; i < N_; i += 256) {
    float a = vtm1b[i];
    float c = vtb[i];
    s_vtm1[i] = a;
    s_vt[i] = c;
    __builtin_nontemporal_store(need_bwd ? a * inv_n : 0.0f, dWRpb + i);
    if (!need_bwd) __builtin_nontemporal_store(0.0f, dWLpb + i);
  }
  if (tid < K_) {
    s_wrx_part[0][tid] = 0.0f;
    s_wrx_part[1][tid] = 0.0f;
    s_wx_part[0][tid] = 0.0f;
    s_wx_part[1][tid] = 0.0f;
  }

  if (!need_fwd && !need_bwd) return;  // block-uniform: skip 256KB of traffic
  __syncthreads();

  if (need_bwd) {
    if (need_fwd)
      heavy_path<true, true>(WL, WR, pb, b, tid, v_next, dWLpb, g, s_vtm1,
                             s_vt, s_wrx_part, s_wx_part, s_wrx, s_wx);
    else
      heavy_path<true, false>(WL, WR, pb, b, tid, v_next, dWLpb, g, s_vtm1,
                              s_vt, s_wrx_part, s_wx_part, s_wrx, s_wx);
  } else {
    heavy_path<false, true>(WL, WR, pb, b, tid, v_next, dWLpb, g, s_vtm1, s_vt,
                            s_wrx_part, s_wx_part, s_wrx, s_wx);
  }
}

extern "C" void kernel_launch(void* const* d_in, const int* in_sizes, int n_in,
                              void* d_out, int out_size, void* d_ws,
                              size_t ws_size, hipStream_t stream) {
  const float* WL = (const float*)d_in[0];
  const float* WR = (const float*)d_in[1];
  const float* v_tm1 = (const float*)d_in[2];
  const float* v_t = (const float*)d_in[3];
  const int* ac_idx = (const int*)d_in[4];
  const int* pc_mask = (const int*)d_in[5];

  float* v_next = (float*)d_out;             // [B,N]
  float* dWL = v_next + (size_t)B_ * N_;     // [P,B,N,1]
  float* dWR = dWL + (size_t)P_ * B_ * N_;   // [P,B,1,N]

  const double rho = 0.999;
  const float g = (float)(0.5 * (log(1.0 + rho) - log(1.0 - rho)) / rho);
  const float inv_n = (float)(1.0 / ((double)N_ * rho * rho));

  hipLaunchKernelGGL(ghu_fused, dim3(P_ * B_), dim3(256), 0, stream, WL, WR,
                     v_tm1, v_t, ac_idx, pc_mask, v_next, dWL, dWR, g, inv_n);
}